// MLPPredictor_26319559590592
// MI455X (gfx1250) — compile-verified
//
#include <hip/hip_runtime.h>
#include <hip/hip_bf16.h>

// MI455X / gfx1250 edge-MLP link predictor.
// score[e] = relu(concat(h[src[e]], h[dst[e]]) @ W1 + b1) @ W2 + b2
//
// 64-edge tiles per 256-thread block (8 wave32 waves):
// - Stage 64x256 f32 concat features into LDS with
//   GLOBAL_LOAD_ASYNC_TO_LDS_B128 (ASYNCcnt path, no VGPR round-trip).
// - Wave w computes hidden columns [16w,16w+16) for 4 M-tiles of 16 edges:
//   64 K-steps of V_WMMA_F32_16X16X4_F32, B fragment loaded once per step
//   and reused across the 4 M-tiles (4x less W1 L2 traffic than 16-edge tiles).
// - Fused bias+ReLU+W2 epilogue, ds_add_f32 reduction to per-edge scores.

typedef float v2f __attribute__((ext_vector_type(2)));
typedef float v8f __attribute__((ext_vector_type(8)));

#define TILE_E    64
#define IN_F      256          // 2 * H_FEATS
#define HID_F     128
#define LDS_PITCH 260          // 256 + 4 pad: rows 16B-aligned, banks spread
#define MT        4            // M-tiles (of 16 edges) per wave

__global__ __launch_bounds__(256)
void edge_mlp_wmma_kernel(const float* __restrict__ h,
                          const int*   __restrict__ src,
                          const int*   __restrict__ dst,
                          const float* __restrict__ W1,   // [256,128] row-major
                          const float* __restrict__ b1,   // [128]
                          const float* __restrict__ W2,   // [128,1]
                          const float* __restrict__ b2,   // [1]
                          float*       __restrict__ out,  // [E]
                          int n_edges)
{
    __shared__ float he[TILE_E][LDS_PITCH];   // concat features, 64 edges
    __shared__ float score[TILE_E];

    const int t    = threadIdx.x;
    const int lane = t & 31;
    const int wave = t >> 5;                  // 0..7
    const long long e_base = (long long)blockIdx.x * TILE_E;

    // Seed per-edge score with b2 (reduced into via LDS atomics below).
    if (t < TILE_E) score[t] = b2[0];

    // ---- Async-stage concat(h[src], h[dst]) tile into LDS -------------
    // 4 threads per edge: parts 0,1 copy the src half (2x64 floats),
    // parts 2,3 copy the dst half. 16 x b128 async copies per thread.
    {
        const int e    = t >> 2;              // 0..63
        const int part = t & 3;               // 0,1 = src half; 2,3 = dst half
        long long ge = e_base + e;
        if (ge >= n_edges) ge = n_edges - 1;  // clamp (E % 64 == 0 anyway)
        const int row  = (part < 2) ? src[ge] : dst[ge];
        const int segF = (part & 1) * 64;     // 64-float segment within half
        unsigned long long ga =
            (unsigned long long)(h + (long long)row * HID_F + segF);
        unsigned lds =
            (unsigned)(size_t)(const void*)&he[e][(part >> 1) * 128 + segF];
#pragma unroll
        for (int i = 0; i < 16; ++i) {
            asm volatile("global_load_async_to_lds_b128 %0, %1, off"
                         :: "v"(lds + 16u * i), "v"(ga + 16ull * i)
                         : "memory");
        }
    }
    asm volatile("s_wait_asynccnt 0x0" ::: "memory");
    __syncthreads();

    // ---- Hidden GEMM: [64 x 256] x [256 x 16] per wave ----------------
    const int n  = wave * 16 + (lane & 15);   // this lane's hidden column
    const int kh = lane >> 4;                 // K half-select (0 or 1)
    const int mA = lane & 15;                 // A-matrix row within M-tile

    const float* bcol = W1 + n;               // column n of W1, stride 128

    v8f acc[MT] = {};
#pragma unroll 2
    for (int k = 0; k < IN_F; k += 4) {
        if ((k & 63) == 0) {
            // keep the W1 K-stream warm (global_prefetch_b8)
            __builtin_prefetch(bcol + (k + 64) * HID_F, 0, 0);
        }
        const int ka = k + 2 * kh;
        // B (4x16 f32): rows ka/ka+1 of W1, col n — loaded once, reused 4x
        v2f bm;
        bm.x = bcol[ka * HID_F];
        bm.y = bcol[(ka + 1) * HID_F];
#pragma unroll
        for (int mt = 0; mt < MT; ++mt) {
            v2f a;                            // A (16x4 f32) per ISA layout
            a.x = he[16 * mt + mA][ka];
            a.y = he[16 * mt + mA][ka + 1];
            acc[mt] = __builtin_amdgcn_wmma_f32_16x16x4_f32(
                          false, a, false, bm, (short)0, acc[mt],
                          false, false);
        }
    }

    // ---- Fused epilogue: relu(acc + b1) * W2, reduce over hidden dim --
    // C layout: VGPR r -> row (r + 8*kh), col (lane & 15).
    const float b1n = b1[n];
    const float w2n = W2[n];
#pragma unroll
    for (int mt = 0; mt < MT; ++mt) {
#pragma unroll
        for (int r = 0; r < 8; ++r) {
            const int m = 16 * mt + r + 8 * kh;   // edge row within tile
            float v = acc[mt][r] + b1n;
            v = v > 0.0f ? v : 0.0f;
            atomicAdd(&score[m], v * w2n);        // ds_add_f32
        }
    }
    __syncthreads();

    if (t < TILE_E) {
        const long long ge = e_base + t;
        if (ge < n_edges) out[ge] = score[t];
    }
}

extern "C" void kernel_launch(void* const* d_in, const int* in_sizes, int n_in,
                              void* d_out, int out_size, void* d_ws, size_t ws_size,
                              hipStream_t stream) {
    const float* h   = (const float*)d_in[0];
    const int*   src = (const int*)d_in[1];
    const int*   dst = (const int*)d_in[2];
    const float* W1  = (const float*)d_in[3];
    const float* b1  = (const float*)d_in[4];
    const float* W2  = (const float*)d_in[5];
    const float* b2  = (const float*)d_in[6];
    float* out = (float*)d_out;

    const int n_edges = in_sizes[1];
    const int grid = (n_edges + TILE_E - 1) / TILE_E;

    edge_mlp_wmma_kernel<<<grid, 256, 0, stream>>>(
        h, src, dst, W1, b1, W2, b2, out, n_edges);
}